// GlobalNode_8650064134241
// MI455X (gfx1250) — compile-verified
//
#include <hip/hip_runtime.h>

#define EMB 128
#define BGRAPH 1024
#define NNODES 500000
#define NEG_SLOPE 0.01f
#define TILES_TOTAL (NNODES / 16)     // 31250 (N divisible by 16)
#define TILES_PER_BLOCK 16            // 256 contiguous nodes per block

typedef __attribute__((ext_vector_type(2))) float v2f;
typedef __attribute__((ext_vector_type(4))) float v4f;
typedef __attribute__((ext_vector_type(8))) float v8f;

__device__ __forceinline__ float leaky(float v) { return v >= 0.f ? v : NEG_SLOPE * v; }

// Order-preserving float->uint encoding so segment-max can use atomicMax(u32).
__device__ __forceinline__ unsigned encf(float f) {
    unsigned u = __float_as_uint(f);
    return (u & 0x80000000u) ? ~u : (u | 0x80000000u);
}
__device__ __forceinline__ float decf(unsigned u) {
    return (u & 0x80000000u) ? __uint_as_float(u ^ 0x80000000u) : __uint_as_float(~u);
}

// ws layout (floats):
// [0, B*128)                xg accumulator
// [B*128, B*129)            denom
// [B*129, B*130)            segmax (as uint bits; 0 == -inf encoding)
// [B*130, B*130 + N)        gate, overwritten with e = exp(gate - m)

__global__ void k_zero(float* ws) {
    int i = blockIdx.x * blockDim.x + threadIdx.x;
    if (i < BGRAPH * 130) ws[i] = 0.f;
}

// One wave per node: coalesced 512B row read, wave-reduce dot, atomic seg-max.
__global__ void k_gate(const float* __restrict__ x, const int* __restrict__ batch,
                       const float* __restrict__ w_mask, const float* __restrict__ b_mask,
                       float* __restrict__ gate, unsigned* __restrict__ segmax) {
    int lane = threadIdx.x & 31;
    int node = blockIdx.x * (blockDim.x >> 5) + (threadIdx.x >> 5);
    if (node >= NNODES) return;
    v4f xv = *(const v4f*)(x + (size_t)node * EMB + lane * 4);
    v4f wv = *(const v4f*)(w_mask + lane * 4);
    float d = xv.x * wv.x + xv.y * wv.y + xv.z * wv.z + xv.w * wv.w;
    for (int off = 16; off > 0; off >>= 1) d += __shfl_xor(d, off, 32);
    if (lane == 0) {
        float g = d + b_mask[0];
        gate[node] = g;
        atomicMax(segmax + batch[node], encf(g));
    }
}

// e = exp(gate - m[batch]); wave-level pre-reduction before atomicAdd(denom).
__global__ void k_expsum(const int* __restrict__ batch, const unsigned* __restrict__ segmax,
                         float* __restrict__ gate, float* __restrict__ denom) {
    int i = blockIdx.x * blockDim.x + threadIdx.x;
    float e = 0.f;
    int b = -1;
    if (i < NNODES) {
        b = batch[i];
        e = expf(gate[i] - decf(segmax[b]));
        gate[i] = e;  // overwrite gate with e
    }
    int first = __shfl(b, 0, 32);
    if (__all(b == first)) {  // sorted batch: almost always uniform
        for (int off = 16; off > 0; off >>= 1) e += __shfl_xor(e, off, 32);
        if ((threadIdx.x & 31) == 0 && b >= 0) atomicAdd(denom + b, e);
    } else if (b >= 0) {
        atomicAdd(denom + b, e);
    }
}

// Main kernel: feat = leaky(x @ w_feat + b_feat) via V_WMMA_F32_16X16X4_F32,
// weighted by alpha, pooled per sorted segment with register accumulators.
//
// w_feat is staged in LDS pre-swizzled into B-fragment order so each lane's
// two K-values per fragment are one contiguous ds_load_b64 with a pure
// immediate offset (kk*2048 bytes) -- no repacking movs in the WMMA loop.
// sWs[((kk*2 + half)*128 + col)*2 + v] = w_feat[4*kk + 2*half + v][col]
__global__ __launch_bounds__(128)
void k_feat(const float* __restrict__ x, const int* __restrict__ batch,
            const float* __restrict__ w_feat, const float* __restrict__ b_feat,
            const float* __restrict__ ebuf, const float* __restrict__ denom,
            float* __restrict__ xg) {
    __shared__ float sWs[EMB * EMB];  // 64 KB: w_feat in B-fragment order
    __shared__ float sX[16 * EMB];    // 8 KB: x tile  [m][k]
    __shared__ float sF[16 * EMB];    // 8 KB: weighted feat tile [m][n]
    __shared__ float sAlpha[16];
    __shared__ int   sBatch[16];

    const int tid = threadIdx.x;
    const int wave = tid >> 5, lane = tid & 31;
    const int half = lane >> 4, lm = lane & 15;

    // Stage + swizzle w_feat once per block (coalesced reads, LDS scatter).
    for (int idx = tid * 4; idx < EMB * EMB; idx += 128 * 4) {
        v4f wv = *(const v4f*)(w_feat + idx);
        int k = idx >> 7;        // source row (K)
        int col = idx & 127;     // first of 4 consecutive columns
        int kk = k >> 2, hf = (k >> 1) & 1, v = k & 1;
        int dst = (((kk * 2 + hf) * 128 + col) << 1) + v;
        sWs[dst]     = wv.x;
        sWs[dst + 2] = wv.y;
        sWs[dst + 4] = wv.z;
        sWs[dst + 6] = wv.w;
    }

    // Each wave owns output columns [16w,16w+16) and [64+16w, 64+16w+16).
    const int col0 = 16 * wave + lm;
    const int col1 = 64 + 16 * wave + lm;
    const float bb0 = b_feat[col0];
    const float bb1 = b_feat[col1];
    // Lane-constant bases into the swizzled B array (elements).
    const int bBase0 = ((half * 128 + col0) << 1);
    const int bBase1 = ((half * 128 + col1) << 1);

    int t0 = blockIdx.x * TILES_PER_BLOCK;
    int t1 = t0 + TILES_PER_BLOCK;
    if (t1 > TILES_TOTAL) t1 = TILES_TOTAL;

    float accReg = 0.f;   // per-thread: column `tid` of current segment's xg row
    int curSeg = -1;

    for (int t = t0; t < t1; ++t) {
        __syncthreads();  // protect sX/sBatch from previous iteration's readers
        const int base = t * 16;
        for (int idx = tid * 4; idx < 16 * EMB; idx += 128 * 4)
            *(v4f*)(sX + idx) = *(const v4f*)(x + (size_t)base * EMB + idx);
        if (tid < 16) {
            int node = base + tid;
            int b = batch[node];
            sBatch[tid] = b;
            sAlpha[tid] = ebuf[node] / denom[b];
        }
        __syncthreads();

        // 16x16 output tiles, K = 128 in 32 steps of 4 (fp32 WMMA).
        v8f c0 = {}; v8f c1 = {};
#pragma unroll
        for (int kk = 0; kk < 32; ++kk) {
            int kb = 4 * kk + 2 * half;
            // A 16x4: lane(<16) M=lm holds K=kb,kb+1; lane(>=16) holds K=kb,kb+1 (=4kk+2,4kk+3)
            v2f a = *(const v2f*)(sX + lm * EMB + kb);
            // B 4x16: single b64 each, immediate offset kk*2048 bytes
            v2f b0 = *(const v2f*)(sWs + bBase0 + kk * 512);
            v2f b1 = *(const v2f*)(sWs + bBase1 + kk * 512);
            c0 = __builtin_amdgcn_wmma_f32_16x16x4_f32(false, a, false, b0, (short)0, c0, false, false);
            c1 = __builtin_amdgcn_wmma_f32_16x16x4_f32(false, a, false, b1, (short)0, c1, false, false);
        }

        float av[8];
#pragma unroll
        for (int r = 0; r < 8; ++r) av[r] = sAlpha[r + 8 * half];
#pragma unroll
        for (int r = 0; r < 8; ++r) {
            int m = r + 8 * half;  // C/D layout: VGPR r, lanes>=16 hold M=r+8
            sF[m * EMB + col0] = leaky(c0[r] + bb0) * av[r];
            sF[m * EMB + col1] = leaky(c1[r] + bb1) * av[r];
        }
        __syncthreads();

        // Sorted-segment pooling: thread `tid` owns column tid; uniform branches.
#pragma unroll
        for (int m = 0; m < 16; ++m) {
            int s = sBatch[m];
            if (s != curSeg) {
                if (curSeg >= 0) atomicAdd(xg + curSeg * EMB + tid, accReg);
                accReg = 0.f;
                curSeg = s;
            }
            accReg += sF[m * EMB + tid];
        }
    }
    if (curSeg >= 0) atomicAdd(xg + curSeg * EMB + tid, accReg);
}

// out = leaky([xg, xg_old] @ w_tr + b_tr) + xg_old   (tiny: 33.5M MACs)
__global__ void k_final(const float* __restrict__ xg, const float* __restrict__ xg_old,
                        const float* __restrict__ w_tr, const float* __restrict__ b_tr,
                        float* __restrict__ out) {
    int i = blockIdx.x * blockDim.x + threadIdx.x;
    if (i >= BGRAPH * EMB) return;
    int row = i / EMB, c = i % EMB;
    const float* xgr = xg + row * EMB;
    const float* xold = xg_old + row * EMB;
    float s = b_tr[c];
#pragma unroll 8
    for (int k = 0; k < EMB; ++k) s += xgr[k] * w_tr[k * EMB + c];
#pragma unroll 8
    for (int k = 0; k < EMB; ++k) s += xold[k] * w_tr[(EMB + k) * EMB + c];
    out[i] = leaky(s) + xold[c];
}

extern "C" void kernel_launch(void* const* d_in, const int* in_sizes, int n_in,
                              void* d_out, int out_size, void* d_ws, size_t ws_size,
                              hipStream_t stream) {
    const float* xg_old = (const float*)d_in[0];
    const float* x      = (const float*)d_in[1];
    const int*   batch  = (const int*)d_in[2];
    const float* w_mask = (const float*)d_in[3];
    const float* b_mask = (const float*)d_in[4];
    const float* w_feat = (const float*)d_in[5];
    const float* b_feat = (const float*)d_in[6];
    const float* w_tr   = (const float*)d_in[7];
    const float* b_tr   = (const float*)d_in[8];
    float* out = (float*)d_out;

    float*    ws     = (float*)d_ws;
    float*    xg     = ws;
    float*    denom  = ws + BGRAPH * EMB;
    unsigned* segmax = (unsigned*)(ws + BGRAPH * EMB + BGRAPH);
    float*    gate   = ws + BGRAPH * EMB + 2 * BGRAPH;

    k_zero<<<(BGRAPH * 130 + 255) / 256, 256, 0, stream>>>(ws);
    k_gate<<<NNODES / 8, 256, 0, stream>>>(x, batch, w_mask, b_mask, gate, segmax);
    k_expsum<<<(NNODES + 255) / 256, 256, 0, stream>>>(batch, segmax, gate, denom);
    int nblocks = (TILES_TOTAL + TILES_PER_BLOCK - 1) / TILES_PER_BLOCK;
    k_feat<<<nblocks, 128, 0, stream>>>(x, batch, w_feat, b_feat, gate, denom, xg);
    k_final<<<(BGRAPH * EMB + 255) / 256, 256, 0, stream>>>(xg, xg_old, w_tr, b_tr, out);
}